// UpResSupp_65798898974754
// MI455X (gfx1250) — compile-verified
//
#include <hip/hip_runtime.h>

typedef __attribute__((ext_vector_type(2))) float v2f;
typedef __attribute__((ext_vector_type(4))) float v4f;
typedef __attribute__((ext_vector_type(8))) float v8f;

#define XU_ELEMS (2 * 32 * 16 * 16 * 16 * 16) /* 4194304 */

// ---------------------------------------------------------------------------
// 4D linear upsample x2 along dims 2..5 : (2,32,8,8,8,8) -> (2,32,16,16,16,16)
// ---------------------------------------------------------------------------
__global__ void upsample4d_kernel(const float* __restrict__ x,
                                  float* __restrict__ xu) {
  int idx = blockIdx.x * blockDim.x + threadIdx.x;
  if (idx >= XU_ELEMS) return;
  int j[4];
  j[3] = idx & 15;
  j[2] = (idx >> 4) & 15;
  j[1] = (idx >> 8) & 15;
  j[0] = (idx >> 12) & 15;
  int bc = idx >> 16;  // b*32 + c
  float fr[4];
  int lo[4], hi[4];
#pragma unroll
  for (int a = 0; a < 4; ++a) {
    float p = (float)j[a] * (7.0f / 15.0f);
    int l = (int)p;  // p >= 0, trunc == floor
    lo[a] = l;
    hi[a] = (l + 1 > 7) ? 7 : (l + 1);
    fr[a] = p - (float)l;
  }
  const float* xb = x + (bc << 12);  // 8^4 = 4096 per (b,c)
  float acc = 0.0f;
#pragma unroll
  for (int m = 0; m < 16; ++m) {
    int c1 = m & 1, c2 = (m >> 1) & 1, c3 = (m >> 2) & 1, c4 = (m >> 3) & 1;
    float w = (c1 ? fr[0] : 1.0f - fr[0]) * (c2 ? fr[1] : 1.0f - fr[1]) *
              (c3 ? fr[2] : 1.0f - fr[2]) * (c4 ? fr[3] : 1.0f - fr[3]);
    int i1 = c1 ? hi[0] : lo[0];
    int i2 = c2 ? hi[1] : lo[1];
    int i3 = c3 ? hi[2] : lo[2];
    int i4 = c4 ? hi[3] : lo[3];
    acc += w * xb[(i1 << 9) + (i2 << 6) + (i3 << 3) + i4];
  }
  xu[idx] = acc;
}

// ---------------------------------------------------------------------------
// Pack conv weights (O=32, I=32, ntap) into per-lane WMMA A-fragment order:
//   wf[((tap*8 + g)*2 + mt)*32 + lane] = float2{ W[o][i0][tap], W[o][i0+1][tap] }
// where o = mt*16 + (lane&15), i0 = 4g + (lane>=16 ? 2 : 0).
// Exactly the V_WMMA_F32_16X16X4_F32 A layout -> kernel A-load is one b64.
// ---------------------------------------------------------------------------
__global__ void pack_w_kernel(const float* __restrict__ w,
                              float* __restrict__ wf, int ntap) {
  int idx = blockIdx.x * blockDim.x + threadIdx.x;
  int total = ntap * 8 * 2 * 32;
  if (idx >= total) return;
  int lane = idx & 31;
  int mt = (idx >> 5) & 1;
  int g = (idx >> 6) & 7;
  int tap = idx >> 9;
  int i0 = 4 * g + ((lane >= 16) ? 2 : 0);
  int o = mt * 16 + (lane & 15);
  wf[idx * 2 + 0] = w[(o * 32 + i0 + 0) * ntap + tap];
  wf[idx * 2 + 1] = w[(o * 32 + i0 + 1) * ntap + tap];
}

// ---------------------------------------------------------------------------
// Implicit-GEMM 4D conv (3^4 taps, pad 1) via V_WMMA_F32_16X16X4_F32.
// Block = 256 thr = 8 waves. Tile per block: 32 out-ch x 256 pixels
// (fixed b,p1; p2 chunk of 2; p3 chunk of 8; p4 full 16).
// Wave w: M-tile mt=w&1; 4 N-tiles at (dp2,dp3) in {0,1}x{0,4}, p3 off nt=w>>1.
//
// LDS layout: channel-pair-major with K-GROUP-ORDERED pairs:
//   physical pair pp = hh*8 + g holds channels {4g+2hh, 4g+2hh+1} interleaved:
//   smv[pp*PSTR + s] = {in[c0][s], in[c0+1][s]}  (v2f, 8B)
// => a wave's 32 B-fragment loads per tap come off TWO base pointers
//    (per-tap sbase, per-lane hh) with small DS offsets. PSTR = 2162
//    (= 2 mod 4) makes the hh half-wave offset 32 mod 64 banks: conflict-free.
// Staging runs per q4-ROW (18 cells): q4 edges are always zero-pad, interior
// 16 cells come from two contiguous 64B-aligned v4f streams.
// HAS_SKIP folds the 1x1 conv (w3 on skip_in) in as one extra K=32 tap.
// ---------------------------------------------------------------------------
#define HALO_NS (3 * 4 * 10 * 18) /* 2160 spatial cells */
#define HALO_ROWS (3 * 4 * 10)    /* 120 q4-rows per pair */
#define PSTR 2162                 /* padded spatial stride per pair (v2f) */
#define NPAIR 16                  /* 32 channels -> 16 interleaved pairs */

template <bool HAS_SKIP>
__global__ __launch_bounds__(256) void conv4d_wmma_kernel(
    const float* __restrict__ in,       // (2,32,16,16,16,16)
    const float* __restrict__ wf,       // packed 81-tap A-fragments
    const float* __restrict__ bias,     // (32)
    const float* __restrict__ skip_in,  // xu (pass 2) or null
    const float* __restrict__ wf3,      // packed 1-tap A-fragments or null
    const float* __restrict__ bias3,    // (32) or null
    float* __restrict__ out) {
  extern __shared__ float smem[];  // 16 pairs * 2*PSTR floats = 276736 B
  v2f* __restrict__ smv = (v2f*)smem;

  int blk = blockIdx.x;  // 512 blocks
  int ck3 = blk & 1;
  int ck2 = (blk >> 1) & 7;
  int p1 = (blk >> 4) & 15;
  int b = blk >> 8;
  int p3b = ck3 * 8;
  int p2b = ck2 * 2;
  int tid = threadIdx.x;

  // ---- stage halo tile by rows: q1:[-1,1], q2:[-1,2], q3:[-1,8], q4:[-1,16]
  for (int row = tid; row < NPAIR * HALO_ROWS; row += 256) {
    int pp = row / HALO_ROWS;
    int r = row - pp * HALO_ROWS;
    int q3 = r % 10;
    int r2 = r / 10;
    int q2 = r2 & 3;
    int q1 = r2 >> 2;
    int g1 = p1 - 1 + q1;
    int g2 = p2b - 1 + q2;
    int g3 = p3b - 1 + q3;
    v2f* dst = smv + pp * PSTR + r * 18;
    v2f z = {0.0f, 0.0f};
    dst[0] = z;   // g4 = -1 : always pad
    dst[17] = z;  // g4 = 16 : always pad
    if (((unsigned)g1 < 16u) & ((unsigned)g2 < 16u) & ((unsigned)g3 < 16u)) {
      int c0 = 4 * (pp & 7) + 2 * (pp >> 3);  // k-group-ordered pair -> chans
      const float* src =
          in + ((b * 32 + c0) << 16) + (g1 << 12) + (g2 << 8) + (g3 << 4);
#pragma unroll
      for (int k = 0; k < 4; ++k) {
        v4f ax = *(const v4f*)(src + 4 * k);          // global_load_b128
        v4f ay = *(const v4f*)(src + 4 * k + 65536);  // next channel
        v2f t;
        t.x = ax.x; t.y = ay.x; dst[1 + 4 * k + 0] = t;
        t.x = ax.y; t.y = ay.y; dst[1 + 4 * k + 1] = t;
        t.x = ax.z; t.y = ay.z; dst[1 + 4 * k + 2] = t;
        t.x = ax.w; t.y = ay.w; dst[1 + 4 * k + 3] = t;
      }
    } else {
#pragma unroll
      for (int q4 = 1; q4 < 17; ++q4) dst[q4] = z;
    }
  }
  __syncthreads();

  int wave = tid >> 5;
  int lane = tid & 31;
  int mt = wave & 1;   // M tile: out-ch base mt*16
  int nt = wave >> 1;  // base p3 offset; tiles at p3 nt / nt+4, p2 +0 / +1
  int lp = lane & 15;  // pixel p4 (B/C rows) or M row (A)
  int hh = lane >> 4;  // half-wave K-pair select

  v8f acc0 = {0.f, 0.f, 0.f, 0.f, 0.f, 0.f, 0.f, 0.f};  // (dp2=0, dp3=0)
  v8f acc1 = acc0;                                       // (dp2=0, dp3=4)
  v8f acc2 = acc0;                                       // (dp2=1, dp3=0)
  v8f acc3 = acc0;                                       // (dp2=1, dp3=4)
  const v2f* __restrict__ wfv = (const v2f*)wf;
  const v2f* __restrict__ smB = smv + (hh ? 8 * PSTR : 0);  // per-lane base

  for (int tap = 0; tap < 81; ++tap) {
    int t4 = tap % 3;
    int tt = tap / 3;
    int t3 = tt % 3;
    tt /= 3;
    int t2 = tt % 3;
    int t1 = tt / 3;
    int sbase = ((t1 * 4 + t2) * 10 + (nt + t3)) * 18 + (lp + t4);
    const v2f* smT = smB + sbase;
    int wbase = tap * 16 + mt;
    if (tap < 80)  // prefetch next tap's A-fragments (4KB, L2-resident)
      __builtin_prefetch(&wfv[(wbase + 16) * 32 + lane], 0, 0);
#pragma unroll
    for (int gh = 0; gh < 2; ++gh) {
      const v2f* smg = smT + gh * (4 * PSTR);  // second base: g = 4..7
#pragma unroll
      for (int g4 = 0; g4 < 4; ++g4) {
        int g = gh * 4 + g4;
        v2f a = wfv[(wbase + g * 2) * 32 + lane];  // one coalesced b64 load
        v2f b00 = smg[g4 * PSTR + 0];    // small-offset ds_load_b64 / 2addr
        v2f b01 = smg[g4 * PSTR + 72];   //   (p3 + 4)
        v2f b10 = smg[g4 * PSTR + 180];  //   (p2 + 1)
        v2f b11 = smg[g4 * PSTR + 252];  //   (both)
        acc0 = __builtin_amdgcn_wmma_f32_16x16x4_f32(
            false, a, false, b00, (short)0, acc0, false, false);
        acc1 = __builtin_amdgcn_wmma_f32_16x16x4_f32(
            false, a, false, b01, (short)0, acc1, false, false);
        acc2 = __builtin_amdgcn_wmma_f32_16x16x4_f32(
            false, a, false, b10, (short)0, acc2, false, false);
        acc3 = __builtin_amdgcn_wmma_f32_16x16x4_f32(
            false, a, false, b11, (short)0, acc3, false, false);
      }
    }
  }

  // output pixel bases: tiles offset +64 (p3+4), +256 (p2+1), +320 (both)
  int spat = (p1 << 12) + (p2b << 8) + ((p3b + nt) << 4) + lp;

  if (HAS_SKIP) {
    const v2f* __restrict__ w3v = (const v2f*)wf3;
#pragma unroll
    for (int g = 0; g < 8; ++g) {
      v2f a = w3v[(g * 2 + mt) * 32 + lane];
      int cb = b * 32 + 4 * g + (hh ? 2 : 0);
      const float* s0 = skip_in + (cb << 16) + spat;
      v2f b00 = {s0[0], s0[65536]};
      v2f b01 = {s0[64], s0[64 + 65536]};
      v2f b10 = {s0[256], s0[256 + 65536]};
      v2f b11 = {s0[320], s0[320 + 65536]};
      acc0 = __builtin_amdgcn_wmma_f32_16x16x4_f32(false, a, false, b00,
                                                   (short)0, acc0, false, false);
      acc1 = __builtin_amdgcn_wmma_f32_16x16x4_f32(false, a, false, b01,
                                                   (short)0, acc1, false, false);
      acc2 = __builtin_amdgcn_wmma_f32_16x16x4_f32(false, a, false, b10,
                                                   (short)0, acc2, false, false);
      acc3 = __builtin_amdgcn_wmma_f32_16x16x4_f32(false, a, false, b11,
                                                   (short)0, acc3, false, false);
    }
  }

  // C/D layout: VGPR r -> M = mt*16 + hh*8 + r, N(pixel p4) = lp
  int obase = mt * 16 + hh * 8;
#pragma unroll
  for (int r = 0; r < 8; ++r) {
    int o = obase + r;
    float bsum = bias[o] + (HAS_SKIP ? bias3[o] : 0.0f);
    float v0 = acc0[r] + bsum;
    float v1 = acc1[r] + bsum;
    float v2 = acc2[r] + bsum;
    float v3 = acc3[r] + bsum;
    v0 = v0 > 0.0f ? v0 : 0.0f;  // ReLU (both passes per reference)
    v1 = v1 > 0.0f ? v1 : 0.0f;
    v2 = v2 > 0.0f ? v2 : 0.0f;
    v3 = v3 > 0.0f ? v3 : 0.0f;
    float* po = out + ((b * 32 + o) << 16) + spat;
    po[0] = v0;
    po[64] = v1;
    po[256] = v2;
    po[320] = v3;
  }
}

// ---------------------------------------------------------------------------
// Workspace layout (floats):
//   xu  : [0, 4194304)        y1 : [4194304, 8388608)
//   wf1 : 82944 floats   wf2 : 82944 floats   wf3 : 1024 floats
// ---------------------------------------------------------------------------
extern "C" void kernel_launch(void* const* d_in, const int* in_sizes, int n_in,
                              void* d_out, int out_size, void* d_ws,
                              size_t ws_size, hipStream_t stream) {
  const float* x = (const float*)d_in[0];
  const float* w1 = (const float*)d_in[1];
  const float* b1 = (const float*)d_in[2];
  const float* w2 = (const float*)d_in[3];
  const float* b2 = (const float*)d_in[4];
  const float* w3 = (const float*)d_in[5];
  const float* b3 = (const float*)d_in[6];
  float* out = (float*)d_out;

  float* ws = (float*)d_ws;
  float* xu = ws;
  float* y1 = ws + 4194304;
  float* wf1 = ws + 8388608;
  float* wf2 = wf1 + 82944;
  float* wf3 = wf2 + 82944;

  upsample4d_kernel<<<XU_ELEMS / 256, 256, 0, stream>>>(x, xu);

  pack_w_kernel<<<(81 * 8 * 2 * 32 + 255) / 256, 256, 0, stream>>>(w1, wf1, 81);
  pack_w_kernel<<<(81 * 8 * 2 * 32 + 255) / 256, 256, 0, stream>>>(w2, wf2, 81);
  pack_w_kernel<<<(1 * 8 * 2 * 32 + 255) / 256, 256, 0, stream>>>(w3, wf3, 1);

  size_t lds_bytes = (size_t)NPAIR * PSTR * sizeof(v2f);  // 276736 B

  conv4d_wmma_kernel<false><<<512, 256, lds_bytes, stream>>>(
      xu, wf1, b1, nullptr, nullptr, nullptr, y1);
  conv4d_wmma_kernel<true><<<512, 256, lds_bytes, stream>>>(
      y1, wf2, b2, xu, wf3, b3, out);
}